// SingleQMSANHead_30442728194141
// MI455X (gfx1250) — compile-verified
//
#include <hip/hip_runtime.h>
#include <math.h>

// MI455X / gfx1250, wave32. f32 WMMA (V_WMMA_F32_16X16X4_F32) for all GEMM
// stages; JS-divergence stage is transcendental-bound VALU with LDS-staged
// query row + coalesced L2-resident key stream.

typedef float v2f __attribute__((ext_vector_type(2)));
typedef float v8f __attribute__((ext_vector_type(8)));

#define EPSQ 1e-10f
#define B_ 4
#define S_ 256
#define E_ 256
#define H_ 128
#define M_ (B_ * S_) // 1024 rows (b,s) flattened

// ---------------------------------------------------------------- reductions
__device__ __forceinline__ float block_reduce_sum(float v, float* sbuf) {
    int t = threadIdx.x;
    sbuf[t] = v;
    __syncthreads();
    for (int off = 128; off >= 1; off >>= 1) {
        if (t < off) sbuf[t] += sbuf[t + off];
        __syncthreads();
    }
    float r = sbuf[0];
    __syncthreads(); // allow buffer reuse
    return r;
}

// fused 3-value reduction: one barrier tree instead of three
__device__ __forceinline__ void block_reduce_sum3(float a, float b, float c,
                                                  float* s0, float* s1, float* s2,
                                                  float& ra, float& rb, float& rc) {
    int t = threadIdx.x;
    s0[t] = a; s1[t] = b; s2[t] = c;
    __syncthreads();
    for (int off = 128; off >= 1; off >>= 1) {
        if (t < off) {
            s0[t] += s0[t + off];
            s1[t] += s1[t + off];
            s2[t] += s2[t + off];
        }
        __syncthreads();
    }
    ra = s0[0]; rb = s1[0]; rc = s2[0];
    __syncthreads();
}

// fused 2-value reduction
__device__ __forceinline__ void block_reduce_sum2(float a, float b,
                                                  float* s0, float* s1,
                                                  float& ra, float& rb) {
    int t = threadIdx.x;
    s0[t] = a; s1[t] = b;
    __syncthreads();
    for (int off = 128; off >= 1; off >>= 1) {
        if (t < off) {
            s0[t] += s0[t + off];
            s1[t] += s1[t + off];
        }
        __syncthreads();
    }
    ra = s0[0]; rb = s1[0];
    __syncthreads();
}

// ------------------------------------------------- Kernel 1: fused x@W GEMMs
// A = x [1024 x 256]; N columns = concat(Wv(128), Wq(256), Wk(256)) = 640.
// One wave per 16x16 output tile; K loop in steps of 4 via WMMA f32 16x16x4.
__global__ void k_gemm_xw(const float* __restrict__ x,
                          const float* __restrict__ Wv,
                          const float* __restrict__ Wq,
                          const float* __restrict__ Wk,
                          float* __restrict__ Vb,
                          float* __restrict__ AQ,
                          float* __restrict__ AK) {
    const int NT_N = (H_ + 2 * E_) / 16; // 40 N-tiles
    int wid  = (blockIdx.x * blockDim.x + threadIdx.x) >> 5; // wave32
    int lane = threadIdx.x & 31;
    int tm = wid / NT_N, tn = wid % NT_N;

    int mrow  = tm * 16 + (lane & 15);      // A-frag: row per lane
    int khalf = (lane < 16) ? 0 : 2;        // K pair 0/1 or 2/3
    int nf    = tn * 16 + (lane & 15);      // global output column

    // pick which weight matrix this column belongs to (uniform per tile)
    const float* wp;
    int col;
    if (nf < H_)            { wp = Wv; col = nf; }
    else if (nf < H_ + E_)  { wp = Wq; col = nf - H_; }
    else                    { wp = Wk; col = nf - H_ - E_; }

    v8f c = {};
    for (int k0 = 0; k0 < E_; k0 += 4) {
        // A 16x4: lane holds {x[row][k], x[row][k+1]} with k = k0+khalf
        v2f a = *(const v2f*)(x + mrow * E_ + k0 + khalf);
        // B 4x16: B[k][n] = W[n][k]; VGPR0/1 = k, k+1 -> one float2 load
        v2f b = *(const v2f*)(wp + col * E_ + k0 + khalf);
        c = __builtin_amdgcn_wmma_f32_16x16x4_f32(false, a, false, b,
                                                  (short)0, c, false, false);
    }

    int rbase = tm * 16 + ((lane < 16) ? 0 : 8);
#pragma unroll
    for (int v = 0; v < 8; ++v) {
        int m = rbase + v;
        float val = c[v];
        if (nf < H_)           Vb[m * H_ + nf] = val;
        else if (nf < H_ + E_) AQ[m * E_ + (nf - H_)] = val;
        else                   AK[m * E_ + (nf - H_ - E_)] = val;
    }
}

// --------------------------------- Kernel 2: probabilities + row entropies
// p = amp^2 / (sum(amp^2) + EPS*||x||^2)  (psi-normalization folded in).
// Writes rho_q [m][e], rho_k transposed [b][e][s], and sp/sq = sum p*log(p+EPS).
__global__ void k_probs(const float* __restrict__ x,
                        const float* __restrict__ AQ,
                        const float* __restrict__ AK,
                        float* __restrict__ RQ,
                        float* __restrict__ RKT,
                        float* __restrict__ SP,
                        float* __restrict__ SQ) {
    __shared__ float s0[256], s1[256], s2[256];
    int m = blockIdx.x;  // 0..1023
    int e = threadIdx.x; // 0..255
    float xv = x[m * E_ + e];
    float aq = AQ[m * E_ + e];
    float ak = AK[m * E_ + e];
    float n2, sumq, sumk;
    block_reduce_sum3(xv * xv, aq * aq, ak * ak, s0, s1, s2, n2, sumq, sumk);
    float nn = sqrtf(n2) + 1e-12f;
    float neps = EPSQ * nn * nn;
    float pq = aq * aq / (sumq + neps);
    float pk = ak * ak / (sumk + neps);
    float sp, sq;
    block_reduce_sum2(pq * __logf(pq + EPSQ), pk * __logf(pk + EPSQ),
                      s0, s1, sp, sq);
    int b = m >> 8, s = m & 255;
    RQ[m * E_ + e] = pq;
    RKT[(b * E_ + e) * S_ + s] = pk; // transposed for coalesced pair loop
    if (e == 0) { SP[m] = sp; SQ[m] = sq; }
}

// ------------------- Kernel 3: JS scores + causal mask + row L1 normalize
// js(i,j) = 0.5*(sp_i + sq_j) - 0.5 * sum_e (P+Q)*log(0.5*(P+Q)+EPS)
// One block per (b,i); thread j computes one pair. Query row staged in LDS
// (broadcast ds reads); key stream lane-coalesced from L2. Scores > 0 so the
// L1 row sum equals the plain sum.
__global__ void k_scores(const float* __restrict__ RQ,
                         const float* __restrict__ RKT,
                         const float* __restrict__ SP,
                         const float* __restrict__ SQ,
                         float* __restrict__ ATTN) {
    __shared__ float qrow[E_];
    __shared__ float sbuf[256];
    int m = blockIdx.x; // b*S + i
    int b = m >> 8, i = m & 255;
    int j = threadIdx.x;

    qrow[j] = RQ[m * E_ + j]; // stage query row once per block
    __syncthreads();

    float score = 0.0f;
    if (j <= i) {
        const float* kT = RKT + b * E_ * S_ + j; // lane-coalesced, stride S_
        float acc = 0.0f;
        for (int e = 0; e < E_; e += 4) {
            __builtin_prefetch(kT + (e + 32) * S_, 0, 0); // cover L2 latency
#pragma unroll
            for (int u = 0; u < 4; ++u) {
                float P = qrow[e + u];      // LDS broadcast
                float Q = kT[(e + u) * S_]; // coalesced global
                float mm = P + Q;
                acc += mm * __logf(0.5f * mm + EPSQ);
            }
        }
        float js = 0.5f * (SP[m] + SQ[b * S_ + j]) - 0.5f * acc;
        score = 1.0f - js;
    }
    float rowsum = block_reduce_sum(score, sbuf);
    float inv = 1.0f / fmaxf(rowsum, 1e-12f);
    ATTN[m * S_ + j] = (j <= i) ? score * inv : 0.0f;
}

// -------------------------------------- Kernel 4: out = attn @ V via WMMA f32
// Per batch: M=S=256, N=H=128, K=S=256. One wave per 16x16 tile (512 tiles).
__global__ void k_av(const float* __restrict__ ATTN,
                     const float* __restrict__ Vb,
                     float* __restrict__ out) {
    int wid  = (blockIdx.x * blockDim.x + threadIdx.x) >> 5; // 0..511
    int lane = threadIdx.x & 31;
    int b  = wid >> 7;        // 4 batches
    int ti = (wid >> 3) & 15; // 16 row tiles
    int tn = wid & 7;         // 8 col tiles

    const float* arow = ATTN + (b * S_ + ti * 16 + (lane & 15)) * S_;
    int khalf = (lane < 16) ? 0 : 2;
    int n = tn * 16 + (lane & 15);
    const float* vcol = Vb + b * S_ * H_ + n;

    v8f c = {};
    for (int k0 = 0; k0 < S_; k0 += 4) {
        v2f a = *(const v2f*)(arow + k0 + khalf);
        v2f bb;
        bb.x = vcol[(k0 + khalf) * H_];     // B VGPR0: row k
        bb.y = vcol[(k0 + khalf + 1) * H_]; // B VGPR1: row k+1
        c = __builtin_amdgcn_wmma_f32_16x16x4_f32(false, a, false, bb,
                                                  (short)0, c, false, false);
    }

    int rbase = b * S_ + ti * 16 + ((lane < 16) ? 0 : 8);
#pragma unroll
    for (int v = 0; v < 8; ++v)
        out[(rbase + v) * H_ + n] = c[v];
}

// ---------------------------------------------------------------- launcher
extern "C" void kernel_launch(void* const* d_in, const int* in_sizes, int n_in,
                              void* d_out, int out_size, void* d_ws,
                              size_t ws_size, hipStream_t stream) {
    const float* x  = (const float*)d_in[0]; // [4,256,256]
    const float* Wv = (const float*)d_in[1]; // [128,256]
    const float* Wq = (const float*)d_in[2]; // [256,256]
    const float* Wk = (const float*)d_in[3]; // [256,256]
    float* out = (float*)d_out;              // [4,256,128]

    float* ws   = (float*)d_ws;
    float* Vb   = ws;             // M_*H_   = 131072
    float* AQ   = Vb + M_ * H_;   // M_*E_   = 262144
    float* AK   = AQ + M_ * E_;   // 262144
    float* RQ   = AK + M_ * E_;   // 262144
    float* RKT  = RQ + M_ * E_;   // 262144
    float* SP   = RKT + M_ * E_;  // 1024
    float* SQ   = SP + M_;        // 1024
    float* ATTN = SQ + M_;        // M_*S_   = 262144  (~5.8 MB total)

    // 64 M-tiles * 40 N-tiles = 2560 waves -> 640 blocks of 4 waves
    k_gemm_xw<<<640, 128, 0, stream>>>(x, Wv, Wq, Wk, Vb, AQ, AK);
    k_probs<<<M_, 256, 0, stream>>>(x, AQ, AK, RQ, RKT, SP, SQ);
    k_scores<<<M_, 256, 0, stream>>>(RQ, RKT, SP, SQ, ATTN);
    // 512 tiles -> 128 blocks of 4 waves
    k_av<<<128, 128, 0, stream>>>(ATTN, Vb, out);
}